// PermutationLayer_86148454023571
// MI455X (gfx1250) — compile-verified
//
#include <hip/hip_runtime.h>
#include <hip/hip_bf16.h>

// ===========================================================================
// Complex-valued MLP over pairwise features (MI455X / gfx1250, wave32).
//
// Precision: reference is complex64 -> V_WMMA_F32_16X16X4_F32 (f32 WMMA),
// no precision loss. Complex GEMM = 4 real WMMA accumulations per K-step:
//   Cre += Are*Wre + Aim*(-Wim) ;  Cim += Are*Wim + Aim*Wre
//
// Data movement: tiles staged global->LDS with GLOBAL_LOAD_ASYNC_TO_LDS_B128
// (ASYNCcnt path, no VGPR round-trip), double-buffered so the async copies of
// slab t+1 overlap the WMMA stream on slab t. Falls back to synchronous
// float4 staging if the async builtin is unavailable.
//
// phase_amplitude_relu(z) = relu(|z|)*e^{i angle z} = z (|z|>=0) -> identity.
// ===========================================================================

typedef __attribute__((ext_vector_type(2))) float v2f;
typedef __attribute__((ext_vector_type(8))) float v8f;
typedef __attribute__((ext_vector_type(4))) int   v4i;

#if defined(__has_builtin)
#if __has_builtin(__builtin_amdgcn_global_load_async_to_lds_b128) && \
    __has_builtin(__builtin_amdgcn_s_wait_asynccnt)
#define USE_ASYNC_LDS 1
#endif
#endif

// builtin signature (from hipcc diagnostic): (v4i AS1*, v4i AS3*, imm, imm)
typedef __attribute__((address_space(1))) v4i gv4i;   // global int4
typedef __attribute__((address_space(3))) v4i sv4i;   // LDS int4

#define LDSTR 20   // LDS row stride (floats): 16B-aligned float4 slots
                   // (20*r+4c mult of 4 floats) and conflict-free b64
                   // fragment reads (20*r mod 64 distinct for r=0..15).

// stage 16 bytes global -> LDS (per lane)
__device__ __forceinline__ void stage16(const float* gp, float* lp)
{
#if USE_ASYNC_LDS
    __builtin_amdgcn_global_load_async_to_lds_b128((gv4i*)gp, (sv4i*)lp, 0, 0);
#else
    *(float4*)lp = *(const float4*)gp;
#endif
}

__device__ __forceinline__ void wait_stage()
{
#if USE_ASYNC_LDS
    __builtin_amdgcn_s_wait_asynccnt(0);   // my wave's async copies landed
#endif
    __syncthreads();                       // everyone's landed / done reading
}

// ---------------------------------------------------------------------------
// Generic complex GEMM:  C[M,N] = A[M,K] @ W[N,K]^T (+ bias[N])
// A, W, C as separate re/im f32 planes. M, N multiples of 64; K multiple 16.
// Block: 256 threads (8 waves), 64x64 block tile; each wave owns two 16x16
// subtiles sharing one column block (W fragments reused). LDS double-buffered.
// ---------------------------------------------------------------------------
__global__ __launch_bounds__(256)
void cgemm_wmma_f32(const float* __restrict__ Are, const float* __restrict__ Aim, int lda,
                    const float* __restrict__ Wre, const float* __restrict__ Wim, int ldw,
                    const float* __restrict__ bre, const float* __restrict__ bim,
                    float* __restrict__ Cre, float* __restrict__ Cim, int ldc,
                    int K)
{
    // [buffer][plane][64 rows * LDSTR]; planes: 0=Are 1=Aim 2=Wre 3=Wim
    __shared__ float smem[2][4][64 * LDSTR];

    const int tid  = threadIdx.x;
    const int wave = tid >> 5;
    const int lane = tid & 31;

    const int tc  = wave & 3;        // column subtile 0..3 (shared by both)
    const int tr0 = wave >> 2;       // row subtile 0..1
    const int tr1 = tr0 + 2;         // row subtile 2..3

    const int rowBase = blockIdx.y * 64;
    const int colBase = blockIdx.x * 64;

    v8f accRe0 = {}; v8f accIm0 = {};
    v8f accRe1 = {}; v8f accIm1 = {};

    // cooperative staging map: 256 threads -> 64 rows x 16 cols per plane
    const int lr = tid >> 2;          // 0..63
    const int lc = (tid & 3) * 4;     // 0,4,8,12

    const int l15  = lane & 15;
    const int half = lane >> 4;       // 0: K {0,1}/M 0..7 ; 1: K {2,3}/M 8..15

    const float* gA0 = Are + (size_t)(rowBase + lr) * lda + lc;
    const float* gA1 = Aim + (size_t)(rowBase + lr) * lda + lc;
    const float* gW0 = Wre + (size_t)(colBase + lr) * ldw + lc;
    const float* gW1 = Wim + (size_t)(colBase + lr) * ldw + lc;
    const int    lofs = lr * LDSTR + lc;

    auto stage_tile = [&](int kb, int buf) {
        stage16(gA0 + kb, &smem[buf][0][lofs]);
        stage16(gA1 + kb, &smem[buf][1][lofs]);
        stage16(gW0 + kb, &smem[buf][2][lofs]);
        stage16(gW1 + kb, &smem[buf][3][lofs]);
    };

    const int ntiles = K >> 4;

    stage_tile(0, 0);
    wait_stage();

    for (int t = 0; t < ntiles; ++t) {
        const int cur = t & 1;
        if (t + 1 < ntiles) stage_tile((t + 1) << 4, cur ^ 1);   // overlap

        const float* sAre = smem[cur][0];
        const float* sAim = smem[cur][1];
        const float* sWre = smem[cur][2];
        const float* sWim = smem[cur][3];

        #pragma unroll
        for (int kk = 0; kk < 4; ++kk) {
            const int kidx  = kk * 4 + half * 2;     // even -> b64-aligned
            const int arow0 = tr0 * 16 + l15;
            const int arow1 = tr1 * 16 + l15;
            const int wrow  = tc  * 16 + l15;

            const v2f aRe0 = *(const v2f*)&sAre[arow0 * LDSTR + kidx];
            const v2f aIm0 = *(const v2f*)&sAim[arow0 * LDSTR + kidx];
            const v2f aRe1 = *(const v2f*)&sAre[arow1 * LDSTR + kidx];
            const v2f aIm1 = *(const v2f*)&sAim[arow1 * LDSTR + kidx];
            const v2f wRe  = *(const v2f*)&sWre[wrow  * LDSTR + kidx];
            const v2f wIm  = *(const v2f*)&sWim[wrow  * LDSTR + kidx];
            const v2f wImN = -wIm;

            accRe0 = __builtin_amdgcn_wmma_f32_16x16x4_f32(false, aRe0, false, wRe,  (short)0, accRe0, false, false);
            accRe0 = __builtin_amdgcn_wmma_f32_16x16x4_f32(false, aIm0, false, wImN, (short)0, accRe0, false, false);
            accIm0 = __builtin_amdgcn_wmma_f32_16x16x4_f32(false, aRe0, false, wIm,  (short)0, accIm0, false, false);
            accIm0 = __builtin_amdgcn_wmma_f32_16x16x4_f32(false, aIm0, false, wRe,  (short)0, accIm0, false, false);

            accRe1 = __builtin_amdgcn_wmma_f32_16x16x4_f32(false, aRe1, false, wRe,  (short)0, accRe1, false, false);
            accRe1 = __builtin_amdgcn_wmma_f32_16x16x4_f32(false, aIm1, false, wImN, (short)0, accRe1, false, false);
            accIm1 = __builtin_amdgcn_wmma_f32_16x16x4_f32(false, aRe1, false, wIm,  (short)0, accIm1, false, false);
            accIm1 = __builtin_amdgcn_wmma_f32_16x16x4_f32(false, aIm1, false, wRe,  (short)0, accIm1, false, false);
        }

        if (t + 1 < ntiles) wait_stage();
    }

    // ---- epilogue: ISA 16x16 f32 D layout + optional bias ----------------
    const int col = colBase + tc * 16 + l15;
    const float brv = bre ? bre[col] : 0.0f;
    const float biv = bim ? bim[col] : 0.0f;
    const int rtop = half * 8;   // VGPR r: M=r (lanes 0-15) / M=8+r (16-31)

    #pragma unroll
    for (int r = 0; r < 8; ++r) {
        const size_t m0 = (size_t)(rowBase + tr0 * 16 + rtop + r);
        Cre[m0 * ldc + col] = accRe0[r] + brv;
        Cim[m0 * ldc + col] = accIm0[r] + biv;
        const size_t m1 = (size_t)(rowBase + tr1 * 16 + rtop + r);
        Cre[m1 * ldc + col] = accRe1[r] + brv;
        Cim[m1 * ldc + col] = accIm1[r] + biv;
    }
}

// ---------------------------------------------------------------------------
// complex64 (interleaved) -> separate re/im planes
// ---------------------------------------------------------------------------
__global__ void deinterleave_c64(const float2* __restrict__ src,
                                 float* __restrict__ re, float* __restrict__ im, int n)
{
    int i = blockIdx.x * 256 + threadIdx.x;
    if (i < n) { float2 v = src[i]; re[i] = v.x; im[i] = v.y; }
}

// ---------------------------------------------------------------------------
// Build Z [4096, 256] for one 64-batch chunk:
//   Z[(bl,i,j), 0:128]   = sum_f x[b,i,f] * Y[(bl,j), k*64+f]   (complex dot)
//   Z[(bl,i,j), 128:192] = ENC[(bl,i), :]
//   Z[(bl,i,j), 192:256] = ENC[(bl,j), :]
// ---------------------------------------------------------------------------
__global__ __launch_bounds__(128)
void build_z(const float* __restrict__ xre, const float* __restrict__ xim,
             const float* __restrict__ YRe, const float* __restrict__ YIm,
             const float* __restrict__ ENCRe, const float* __restrict__ ENCIm,
             float* __restrict__ ZRe, float* __restrict__ ZIm, int b0)
{
    __shared__ float sxr[64];
    __shared__ float sxi[64];

    const int r  = blockIdx.x;                 // 0..4095
    const int bl = r >> 6;
    const int i  = (r >> 3) & 7;
    const int j  = r & 7;
    const int t  = threadIdx.x;                // 0..127

    const size_t growi = (size_t)((b0 + bl) * 8 + i) * 64;
    if (t < 64) sxr[t] = xre[growi + t];
    else        sxi[t - 64] = xim[growi + (t - 64)];
    __syncthreads();

    const size_t ybase = (size_t)(bl * 8 + j) * 8192 + (size_t)t * 64;
    const float* yr = YRe + ybase;
    const float* yi = YIm + ybase;

    float acr = 0.0f, aci = 0.0f;
    #pragma unroll 8
    for (int f = 0; f < 64; ++f) {
        const float xr = sxr[f], xi2 = sxi[f];
        const float Yr = yr[f],  Yi  = yi[f];
        acr = fmaf(xr, Yr, fmaf(-xi2, Yi, acr));
        aci = fmaf(xr, Yi, fmaf( xi2, Yr, aci));
    }

    const size_t zb = (size_t)r * 256;
    ZRe[zb + t] = acr;
    ZIm[zb + t] = aci;

    const size_t li = (size_t)(bl * 8 + i) * 64;
    const size_t lj = (size_t)(bl * 8 + j) * 64;
    float er, ei;
    if (t < 64) { er = ENCRe[li + t];        ei = ENCIm[li + t]; }
    else        { er = ENCRe[lj + (t - 64)]; ei = ENCIm[lj + (t - 64)]; }
    ZRe[zb + 128 + t] = er;
    ZIm[zb + 128 + t] = ei;
}

// ---------------------------------------------------------------------------
// Mean over j (axis=2), write interleaved complex64 output.
// ---------------------------------------------------------------------------
__global__ __launch_bounds__(256)
void reduce_mean_j(const float* __restrict__ ORe, const float* __restrict__ OIm,
                   float2* __restrict__ out, int b0)
{
    const int row = blockIdx.x;        // 0..511 -> (bl, i)
    const int k   = threadIdx.x;       // 0..255
    const int bl  = row >> 3;
    const int i   = row & 7;

    float sr = 0.0f, si = 0.0f;
    #pragma unroll
    for (int j = 0; j < 8; ++j) {
        const size_t idx = (size_t)(bl * 64 + i * 8 + j) * 256 + k;
        sr += ORe[idx];
        si += OIm[idx];
    }
    const size_t o = (size_t)((b0 + bl) * 8 + i) * 256 + k;
    out[o] = make_float2(sr * 0.125f, si * 0.125f);
}

// ===========================================================================
extern "C" void kernel_launch(void* const* d_in, const int* in_sizes, int n_in,
                              void* d_out, int out_size, void* d_ws, size_t ws_size,
                              hipStream_t stream)
{
    (void)in_sizes; (void)n_in; (void)out_size; (void)ws_size;

    const float*  xre  = (const float*) d_in[0];   // [2048, 64]
    const float*  xim  = (const float*) d_in[1];
    const float2* Wo   = (const float2*)d_in[2];   // [128,64,64] c64 -> [8192,64]
    const float2* Wp   = (const float2*)d_in[3];   // [64,64]
    const float2* Wh1  = (const float2*)d_in[4];   // [512,256]
    const float2* bh1  = (const float2*)d_in[5];   // [512]
    const float2* Wh2  = (const float2*)d_in[6];   // [512,512]
    const float2* bh2  = (const float2*)d_in[7];   // [512]
    const float2* Wout = (const float2*)d_in[8];   // [256,512]
    const float2* bout = (const float2*)d_in[9];   // [256]

    float* ws = (float*)d_ws;
    size_t off = 0;
    auto alloc = [&](size_t n) { float* p = ws + off; off += n; return p; };

    // weight planes (persistent across chunks)
    const int NWO = 8192 * 64, NWP = 64 * 64, NW1 = 512 * 256, NB1 = 512;
    const int NW2 = 512 * 512, NB2 = 512, NWOUT = 256 * 512, NBOUT = 256;
    float *WoRe = alloc(NWO),  *WoIm = alloc(NWO);
    float *WpRe = alloc(NWP),  *WpIm = alloc(NWP);
    float *W1Re = alloc(NW1),  *W1Im = alloc(NW1);
    float *b1Re = alloc(NB1),  *b1Im = alloc(NB1);
    float *W2Re = alloc(NW2),  *W2Im = alloc(NW2);
    float *b2Re = alloc(NB2),  *b2Im = alloc(NB2);
    float *WuRe = alloc(NWOUT),*WuIm = alloc(NWOUT);
    float *buRe = alloc(NBOUT),*buIm = alloc(NBOUT);

    // per-chunk buffers (chunk = 64 batches -> 512 x-rows, 4096 pair-rows)
    const int NENC = 512 * 64, NY = 512 * 8192, NZ = 4096 * 256;
    const int NH = 4096 * 512, NO = 4096 * 256;
    float *ENCRe = alloc(NENC), *ENCIm = alloc(NENC);
    float *YRe   = alloc(NY),   *YIm   = alloc(NY);
    float *ZRe   = alloc(NZ),   *ZIm   = alloc(NZ);
    float *H1Re  = alloc(NH),   *H1Im  = alloc(NH);
    float *H2Re  = alloc(NH),   *H2Im  = alloc(NH);
    float *ORe   = alloc(NO),   *OIm   = alloc(NO);

    // ---- de-interleave weights once -------------------------------------
    auto dil = [&](const float2* s, float* re, float* im, int n) {
        deinterleave_c64<<<(n + 255) / 256, 256, 0, stream>>>(s, re, im, n);
    };
    dil(Wo,   WoRe, WoIm, NWO);
    dil(Wp,   WpRe, WpIm, NWP);
    dil(Wh1,  W1Re, W1Im, NW1);
    dil(bh1,  b1Re, b1Im, NB1);
    dil(Wh2,  W2Re, W2Im, NW2);
    dil(bh2,  b2Re, b2Im, NB2);
    dil(Wout, WuRe, WuIm, NWOUT);
    dil(bout, buRe, buIm, NBOUT);

    // ---- pipeline, chunked over 4 groups of 64 batches ------------------
    for (int c = 0; c < 4; ++c) {
        const int b0 = c * 64;
        const float* xr_c = xre + (size_t)b0 * 512;   // 64 batches * 8 * 64
        const float* xi_c = xim + (size_t)b0 * 512;

        // Y[(b,j),(k,f)] = X @ WoFlat^T : M=512, N=8192, K=64
        cgemm_wmma_f32<<<dim3(8192 / 64, 512 / 64), 256, 0, stream>>>(
            xr_c, xi_c, 64, WoRe, WoIm, 64, nullptr, nullptr, YRe, YIm, 8192, 64);

        // ENC = X @ Wp^T : M=512, N=64, K=64
        cgemm_wmma_f32<<<dim3(1, 512 / 64), 256, 0, stream>>>(
            xr_c, xi_c, 64, WpRe, WpIm, 64, nullptr, nullptr, ENCRe, ENCIm, 64, 64);

        // Z = [enc_outer | enc_i | enc_j] : [4096, 256]
        build_z<<<4096, 128, 0, stream>>>(xre, xim, YRe, YIm, ENCRe, ENCIm,
                                          ZRe, ZIm, b0);

        // H1 = Z @ Wh1^T + bh1 : M=4096, N=512, K=256  (activation = identity)
        cgemm_wmma_f32<<<dim3(512 / 64, 4096 / 64), 256, 0, stream>>>(
            ZRe, ZIm, 256, W1Re, W1Im, 256, b1Re, b1Im, H1Re, H1Im, 512, 256);

        // H2 = H1 @ Wh2^T + bh2 : M=4096, N=512, K=512
        cgemm_wmma_f32<<<dim3(512 / 64, 4096 / 64), 256, 0, stream>>>(
            H1Re, H1Im, 512, W2Re, W2Im, 512, b2Re, b2Im, H2Re, H2Im, 512, 512);

        // O = H2 @ Wout^T + bout : M=4096, N=256, K=512
        cgemm_wmma_f32<<<dim3(256 / 64, 4096 / 64), 256, 0, stream>>>(
            H2Re, H2Im, 512, WuRe, WuIm, 512, buRe, buIm, ORe, OIm, 256, 512);

        // mean over j -> interleaved complex64 output
        reduce_mean_j<<<512, 256, 0, stream>>>(ORe, OIm, (float2*)d_out, b0);
    }
}